// ProjectionNetwork_37589553774839
// MI455X (gfx1250) — compile-verified
//
#include <hip/hip_runtime.h>
#include <math.h>

#define DIM   512
#define NCOLS 16384

typedef __attribute__((ext_vector_type(2))) float v2f;
typedef __attribute__((ext_vector_type(4))) float v4f;
typedef __attribute__((ext_vector_type(8))) float v8f;

// ---------------------------------------------------------------------------
// Wave32 / block reductions
// ---------------------------------------------------------------------------
__device__ __forceinline__ float wave_reduce_add(float v) {
#pragma unroll
  for (int off = 16; off > 0; off >>= 1)
    v += __shfl_down(v, off, 32);
  return v;
}

__device__ __forceinline__ float block_reduce_add(float v, float* red, int tid) {
  const int lane = tid & 31;
  const int wv   = tid >> 5;          // 16 waves for 512 threads
  v = wave_reduce_add(v);
  __syncthreads();                    // protect red[] from previous use
  if (lane == 0) red[wv] = v;
  __syncthreads();
  float s = 0.f;
#pragma unroll
  for (int k = 0; k < 16; ++k) s += red[k];
  return s;
}

// ---------------------------------------------------------------------------
// Kernel 1: orthonormalize columns of W; emit V (= U^T, row-major: row i is the
// i-th orthonormal vector, contiguous) and Dsig = sigmoid(D_param).
// Single workgroup, 512 threads = 16 wave32 waves; classical one-shot
// projection per column so the 16 waves compute all projection coefficients
// concurrently.
// ---------------------------------------------------------------------------
__global__ __launch_bounds__(512) void gs_kernel(const float* __restrict__ W,
                                                 const float* __restrict__ Dp,
                                                 float* __restrict__ V,
                                                 float* __restrict__ Dsig) {
  __shared__ float wi[DIM];
  __shared__ float c[DIM];
  __shared__ float red[16];
  const int tid  = threadIdx.x;
  const int lane = tid & 31;
  const int wv   = tid >> 5;

  Dsig[tid] = 1.0f / (1.0f + __expf(-Dp[tid]));

  // column 0
  float w0 = W[tid * DIM + 0];
  float ss = block_reduce_add(w0 * w0, red, tid);
  V[tid] = w0 * rsqrtf(ss);
  __threadfence();
  __syncthreads();

  for (int i = 1; i < DIM; ++i) {
    wi[tid] = W[tid * DIM + i];
    __syncthreads();

    // c[j] = <V[j,:], w_i>  — wave wv handles j = wv, wv+16, ...
    for (int j = wv; j < i; j += 16) {
      float p = 0.f;
#pragma unroll 4
      for (int r = lane; r < DIM; r += 32)
        p += V[j * DIM + r] * wi[r];
      p = wave_reduce_add(p);
      if (lane == 0) c[j] = p;
    }
    __syncthreads();

    // v[r] = w_i[r] - sum_j c[j] * V[j][r]   (coalesced: tid indexes r)
    float v = wi[tid];
    for (int j = 0; j < i; ++j)
      v -= c[j] * V[j * DIM + tid];

    float n2 = block_reduce_add(v * v, red, tid);
    V[i * DIM + tid] = v * rsqrtf(n2);
    __threadfence();
    __syncthreads();
  }
}

// ---------------------------------------------------------------------------
// Kernel 2: P = V^T * diag(Dsig) * V   (512x512, symmetric).
// One wave32 per 16x16 tile, K stepped by 4 with V_WMMA_F32_16X16X4_F32.
// A[m][k] = Dsig[k]*V[k][m]  and  B[k][n] = V[k][n]  — both contiguous rows.
// Output is stored FRAGMENT-PACKED for the px GEMM: even/odd first-index rows
// interleaved, Ppk[(row>>1)*2*DIM + col*2 + (row&1)], so an A-fragment pair
// (P[k][m], P[k+1][m]) is a single aligned 8-byte load.
// ---------------------------------------------------------------------------
__global__ __launch_bounds__(256) void pmat_kernel(const float* __restrict__ V,
                                                   const float* __restrict__ Dsig,
                                                   float* __restrict__ Ppk) {
  const int tid = threadIdx.x;
  const int lane = tid & 31;
  const int wv   = tid >> 5;
  const int hh   = lane >> 4;        // half-wave select
  const int l16  = lane & 15;
  const int m0 = blockIdx.x * 128 + wv * 16;
  const int n0 = blockIdx.y * 16;

  v8f acc = {};
  for (int k = 0; k < DIM; k += 4) {
    const int ka = k + 2 * hh;       // VGPR0 K, VGPR1 is ka+1
    v2f a, b;
    a.x = Dsig[ka]     * V[ka * DIM + m0 + l16];
    a.y = Dsig[ka + 1] * V[(ka + 1) * DIM + m0 + l16];
    b.x = V[ka * DIM + n0 + l16];
    b.y = V[(ka + 1) * DIM + n0 + l16];
    acc = __builtin_amdgcn_wmma_f32_16x16x4_f32(false, a, false, b,
                                                (short)0, acc, false, false);
  }
#pragma unroll
  for (int r = 0; r < 8; ++r) {
    const int row = m0 + r + 8 * hh; // C/D layout: VGPR r -> M = r + 8*half
    const int col = n0 + l16;
    Ppk[(size_t)(row >> 1) * (2 * DIM) + col * 2 + (row & 1)] = acc[r];
  }
}

// ---------------------------------------------------------------------------
// Kernel 3: out = P @ x  (512 x 16384). Compute-bound fp32 GEMM on WMMA.
// Workgroup = 8 waves. Wave tile = 16(M) x 64(N): 4 accumulators so each
// A-fragment feeds 4 WMMAs; A pairs are single b64 loads from packed P
// (L2-resident, P symmetric -> P[k][m], m contiguous across lanes).
// x tiles (32 x 64) are DOUBLE-BUFFERED into LDS with CDNA5 async copies
// (GLOBAL_LOAD_ASYNC_TO_LDS_B128, tracked by ASYNCcnt): issue chunk i+1,
// compute chunk i, s_wait_asynccnt 0 + barrier before consuming.
// LDS row stride padded to 80 floats so the two half-waves of a B-fragment
// (rows ka / ka+2) land on disjoint bank ranges.
// ---------------------------------------------------------------------------
#define KC    32                     // K-chunk
#define XPAD  80                     // LDS row stride (floats), conflict-free

__global__ __launch_bounds__(256) void px_kernel(const float* __restrict__ Ppk,
                                                 const float* __restrict__ x,
                                                 float* __restrict__ out) {
  __shared__ __align__(16) float xs[2][KC][XPAD];
  const int tid = threadIdx.x;
  const int lane = tid & 31;
  const int wv   = tid >> 5;
  const int hh   = lane >> 4;
  const int l16  = lane & 15;
  const int m0 = blockIdx.x * 128 + wv * 16;   // wave's M rows
  const int n0 = blockIdx.y * 64;              // workgroup's N strip
  const int sr = tid >> 4;                     // stage row   0..15
  const int sc = (tid & 15) * 4;               // stage col4  0..60

  // async memory->LDS stage of one 32x64 chunk (2 x b128 per thread)
  auto stage = [&](int buf, int k0) {
#pragma unroll
    for (int p = 0; p < KC / 16; ++p) {
      const int r = sr + 16 * p;
      unsigned lds = (unsigned)(uintptr_t)&xs[buf][r][sc];
      unsigned long long ga =
          (unsigned long long)(uintptr_t)&x[(size_t)(k0 + r) * NCOLS + n0 + sc];
      asm volatile("global_load_async_to_lds_b128 %0, %1, off"
                   :: "v"(lds), "v"(ga) : "memory");
    }
  };

  stage(0, 0);

  v8f acc0 = {}, acc1 = {}, acc2 = {}, acc3 = {};

  for (int k0 = 0; k0 < DIM; k0 += KC) {
    const int buf = (k0 / KC) & 1;
    asm volatile("s_wait_asynccnt 0x0" ::: "memory");  // my copies done
    __syncthreads();                                   // everyone's published
    if (k0 + KC < DIM) stage(buf ^ 1, k0 + KC);        // overlap next chunk

#pragma unroll
    for (int kk = 0; kk < KC; kk += 4) {
      const int ka = kk + 2 * hh;
      // packed-P A fragment: one aligned 8B load per lane
      v2f a = *(const v2f*)&Ppk[(size_t)((k0 + ka) >> 1) * (2 * DIM) +
                                2 * (m0 + l16)];
      v2f b0, b1, b2, b3;
      b0.x = xs[buf][ka][ 0 + l16]; b0.y = xs[buf][ka + 1][ 0 + l16];
      b1.x = xs[buf][ka][16 + l16]; b1.y = xs[buf][ka + 1][16 + l16];
      b2.x = xs[buf][ka][32 + l16]; b2.y = xs[buf][ka + 1][32 + l16];
      b3.x = xs[buf][ka][48 + l16]; b3.y = xs[buf][ka + 1][48 + l16];
      acc0 = __builtin_amdgcn_wmma_f32_16x16x4_f32(false, a, false, b0,
                                                   (short)0, acc0, false, false);
      acc1 = __builtin_amdgcn_wmma_f32_16x16x4_f32(false, a, false, b1,
                                                   (short)0, acc1, false, false);
      acc2 = __builtin_amdgcn_wmma_f32_16x16x4_f32(false, a, false, b2,
                                                   (short)0, acc2, false, false);
      acc3 = __builtin_amdgcn_wmma_f32_16x16x4_f32(false, a, false, b3,
                                                   (short)0, acc3, false, false);
    }
  }

#pragma unroll
  for (int r = 0; r < 8; ++r) {
    const size_t row = (size_t)(m0 + r + 8 * hh) * NCOLS;
    out[row + n0 +  0 + l16] = acc0[r];
    out[row + n0 + 16 + l16] = acc1[r];
    out[row + n0 + 32 + l16] = acc2[r];
    out[row + n0 + 48 + l16] = acc3[r];
  }
}

// ---------------------------------------------------------------------------
extern "C" void kernel_launch(void* const* d_in, const int* in_sizes, int n_in,
                              void* d_out, int out_size, void* d_ws, size_t ws_size,
                              hipStream_t stream) {
  const float* x  = (const float*)d_in[0];   // 512 x 16384
  const float* W  = (const float*)d_in[1];   // 512 x 512
  const float* Dp = (const float*)d_in[2];   // 512
  float* out = (float*)d_out;                // 512 x 16384

  float* V    = (float*)d_ws;                // 512*512 floats (U^T, row-major)
  float* Ppk  = V + DIM * DIM;               // 512*512 floats (packed P)
  float* Dsig = Ppk + DIM * DIM;             // 512 floats

  gs_kernel<<<1, 512, 0, stream>>>(W, Dp, V, Dsig);
  pmat_kernel<<<dim3(DIM / 128, DIM / 16), 256, 0, stream>>>(V, Dsig, Ppk);
  px_kernel<<<dim3(DIM / 128, NCOLS / 64), 256, 0, stream>>>(Ppk, x, out);
}